// SemanticSmoothing_29145648071144
// MI455X (gfx1250) — compile-verified
//
#include <hip/hip_runtime.h>
#include <hip/hip_bf16.h>
#include <stdint.h>

// Morphological opening (10x10 flat erosion then dilation, SAME padding) on
// NHWC [16,512,512,8] fp32. Fully fused, separable, LDS-tiled, with CDNA5
// async global->LDS copies (ASYNCcnt / s_wait_asynccnt).

#define N_  16
#define H_  512
#define W_  512
#define C_  8
#define K_  10
#define PADLO 4   // SAME for k=10: lo=4, hi=5
#define HALO1 4   // halo on the low side per stage
#define HT  16    // output tile height
#define WT  16    // output tile width
#define INH (HT + 18)   // 34: input rows needed (two stages of 9 taps extra)
#define INW (WT + 18)   // 34
#define RW  (WT + 9)    // 25: row-min cols / eroded cols
#define EH  (HT + 9)    // 25: eroded rows / row-max rows

__device__ __forceinline__ float4 fmin4(float4 a, float4 b) {
    return make_float4(fminf(a.x, b.x), fminf(a.y, b.y),
                       fminf(a.z, b.z), fminf(a.w, b.w));
}
__device__ __forceinline__ float4 fmax4(float4 a, float4 b) {
    return make_float4(fmaxf(a.x, b.x), fmaxf(a.y, b.y),
                       fmaxf(a.z, b.z), fmaxf(a.w, b.w));
}

__global__ __launch_bounds__(256)
void SemanticSmoothing_opening_kernel(const float* __restrict__ in,
                                      float* __restrict__ out) {
    // bufA: input tile (34x34x8), later reused for eroded tile (25x25x8)
    // bufB: row-min tile (34x25x8), later reused for row-max tile (25x16x8)
    __shared__ __align__(16) float bufA[INH * INW * C_];   // 36,992 B
    __shared__ __align__(16) float bufB[INH * RW  * C_];   // 27,200 B

    const int tid = threadIdx.x;
    const int w0 = blockIdx.x * WT;
    const int h0 = blockIdx.y * HT;
    const int n  = blockIdx.z;

    const uint64_t sbase = (uint64_t)in;
    const float INF = __builtin_huge_valf();

    // ---- Stage 0: async-load input tile (halo 8 low / 10 high), +inf pad ----
    // Each unit is one float4 (half of an 8-channel pixel).
    for (int i = tid; i < INH * INW * 2; i += 256) {
        const int pix  = i >> 1;
        const int half = i & 1;
        const int r = pix / INW;
        const int c = pix - r * INW;
        const int gh = h0 - (2 * HALO1) + r;   // h0 - 8 + r
        const int gw = w0 - (2 * HALO1) + c;
        float* ldsp = &bufA[(pix << 3) + (half << 2)];
        if ((unsigned)gh < (unsigned)H_ && (unsigned)gw < (unsigned)W_) {
            // byte offset into the tensor: ((n*512+gh)*512+gw)*8 floats
            const unsigned off =
                ((((unsigned)(n * H_ + gh) << 9) + (unsigned)gw) << 5) +
                ((unsigned)half << 4);
            const unsigned lds = (unsigned)(uintptr_t)ldsp;  // low 32b = LDS offset
            asm volatile("global_load_async_to_lds_b128 %0, %1, %2"
                         :
                         : "v"(lds), "v"(off), "s"(sbase)
                         : "memory");
        } else {
            *(float4*)ldsp = make_float4(INF, INF, INF, INF);  // erosion identity
        }
    }
    asm volatile("s_wait_asynccnt 0x0" ::: "memory");
    __syncthreads();

    // ---- Stage A: row-min along W: bufB[r][c] = min_{dw=0..9} bufA[r][c+dw] ----
    for (int i = tid; i < INH * RW * 2; i += 256) {
        const int pix  = i >> 1;
        const int half = i & 1;
        const int r = pix / RW;
        const int c = pix - r * RW;
        const float4* src = (const float4*)&bufA[((r * INW + c) << 3) + (half << 2)];
        float4 m = src[0];
        #pragma unroll
        for (int dw = 1; dw < K_; ++dw) m = fmin4(m, src[dw * 2]);
        *(float4*)&bufB[((r * RW + c) << 3) + (half << 2)] = m;
    }
    __syncthreads();

    // ---- Stage B: col-min -> eroded (25x25x8), written over bufA.
    //      Positions outside the image become -inf (dilation padding). ----
    for (int i = tid; i < EH * RW * 2; i += 256) {
        const int pix  = i >> 1;
        const int half = i & 1;
        const int r = pix / RW;
        const int c = pix - r * RW;
        const int he = h0 - HALO1 + r;
        const int we = w0 - HALO1 + c;
        float4 m;
        if ((unsigned)he < (unsigned)H_ && (unsigned)we < (unsigned)W_) {
            const float4* src = (const float4*)&bufB[((r * RW + c) << 3) + (half << 2)];
            m = src[0];
            #pragma unroll
            for (int dh = 1; dh < K_; ++dh) m = fmin4(m, src[dh * RW * 2]);
        } else {
            m = make_float4(-INF, -INF, -INF, -INF);
        }
        *(float4*)&bufA[((r * RW + c) << 3) + (half << 2)] = m;
    }
    __syncthreads();

    // ---- Stage C: row-max along W: bufB[r][c] = max_{dw} eroded[r][c+dw] ----
    for (int i = tid; i < EH * WT * 2; i += 256) {
        const int pix  = i >> 1;
        const int half = i & 1;
        const int r = pix / WT;
        const int c = pix - r * WT;
        const float4* src = (const float4*)&bufA[((r * RW + c) << 3) + (half << 2)];
        float4 m = src[0];
        #pragma unroll
        for (int dw = 1; dw < K_; ++dw) m = fmax4(m, src[dw * 2]);
        *(float4*)&bufB[((r * WT + c) << 3) + (half << 2)] = m;
    }
    __syncthreads();

    // ---- Stage D: col-max -> output (x WEIGHT=1.0), b128 stores ----
    for (int i = tid; i < HT * WT * 2; i += 256) {
        const int pix  = i >> 1;
        const int half = i & 1;
        const int r = pix / WT;   // ty
        const int c = pix - r * WT;   // tx
        const float4* src = (const float4*)&bufB[((r * WT + c) << 3) + (half << 2)];
        float4 m = src[0];
        #pragma unroll
        for (int dh = 1; dh < K_; ++dh) m = fmax4(m, src[dh * WT * 2]);
        const int gidx = (((n * H_ + (h0 + r)) * W_) + (w0 + c)) * C_ + (half << 2);
        *(float4*)&out[gidx] = m;   // WEIGHT == 1.0
    }
}

extern "C" void kernel_launch(void* const* d_in, const int* in_sizes, int n_in,
                              void* d_out, int out_size, void* d_ws, size_t ws_size,
                              hipStream_t stream) {
    (void)in_sizes; (void)n_in; (void)out_size; (void)d_ws; (void)ws_size;
    const float* in = (const float*)d_in[0];
    float* out = (float*)d_out;
    dim3 grid(W_ / WT, H_ / HT, N_);   // (32, 32, 16)
    dim3 block(256);
    SemanticSmoothing_opening_kernel<<<grid, block, 0, stream>>>(in, out);
}